// dis_conv_52243982189252
// MI455X (gfx1250) — compile-verified
//
#include <hip/hip_runtime.h>
#include <stdint.h>

typedef __attribute__((ext_vector_type(2))) float v2f;
typedef __attribute__((ext_vector_type(8))) float v8f;
typedef __attribute__((ext_vector_type(4))) unsigned int v4u;
typedef __attribute__((ext_vector_type(8))) int v8i;
typedef __attribute__((ext_vector_type(4))) int v4i;

#define HH 1024
#define WW 2048
#define HI 1022   // H-2
#define WJ 1022   // W/2-2

// k = c*9 + p*3 + q, k < 28. Decompose with shift tricks (valid for k<32).
__device__ __forceinline__ void kdec(int k, int& c, int& p, int& q) {
    int t = (k * 11) >> 5;      // k/3
    q = k - 3 * t;
    c = (t * 11) >> 5;          // t/3
    p = t - 3 * c;
}

// A-matrix gather: G(j,k) = x[b, c, clamp(i+p+bn[i+p, j1+q]), q]  (from LDS)
__device__ __forceinline__ float gatherA(const int* __restrict__ bnl,
                                         const float* __restrict__ xs,
                                         int i, int j, int k) {
    if (k >= 27) return 0.0f;           // K padding 27 -> 28
    int c, p, q; kdec(k, c, p, q);
    int j1 = j & 255;                   // == (j>=256 ? j%256 : j) for j<1024
    int bnv = bnl[p * 260 + j1 + q];
    int r = i + p + bnv;
    r = (r < 0) ? 0 : (r > HH - 1 ? HH - 1 : r);   // jnp index clamping
    return xs[(c * HH + r) * 3 + q];
}

// B-matrix element: weight^T[k][n], zero-padded for n>=3 / k>=27
__device__ __forceinline__ float wB(const float* __restrict__ w, int n, int k) {
    if (n >= 3 || k >= 27) return 0.0f;
    int c, p, q; kdec(k, c, p, q);
    return w[((n * 3 + c) * 3 + p) * 3 + q];
}

__global__ __launch_bounds__(256) void dis_conv_52243982189252_kernel(
        const float* __restrict__ x, const float* __restrict__ weight,
        const int* __restrict__ bn, float* __restrict__ out) {
    const int i   = blockIdx.x;          // output row
    const int b   = blockIdx.y;          // batch
    const int tid = threadIdx.x;
    float* outB = out + (size_t)b * 3 * HH * WW;

    if (i >= HI) {                        // bottom 2 rows: all zero
        #pragma unroll
        for (int o = 0; o < 3; ++o) {
            float4* dst = (float4*)(outB + ((size_t)o * HH + i) * WW);
            for (int col = tid; col < WW / 4; col += 256)
                dst[col] = make_float4(0.f, 0.f, 0.f, 0.f);
        }
        return;
    }

    __shared__ float xslab[3 * HH * 3];   // x[b][c][r][q], q in [0,3): 36 KB
    __shared__ int   bnl[3 * 260];        // bn rows i..i+2, cols 0..257
    __shared__ float trans[8][96];        // per-wave D-tile transpose buffer

    const int lane = tid & 31;
    const int wv   = tid >> 5;

    // ---- TDM: DMA x[b, 0:3, 0:1024, 0:3] -> xslab (wave 0 only, scalar branch)
    if (__builtin_amdgcn_readfirstlane(wv) == 0) {
        uint64_t ga = (uint64_t)(uintptr_t)(x + (size_t)b * 3 * HH * WW);
        uint32_t la = (uint32_t)(uintptr_t)xslab;   // low 32 bits = LDS offset
        v4u g0; v8i g1; v4i g2, g3; v8i g4;
        g0[0] = 1u;                                        // count=1, user mode
        g0[1] = la;                                        // lds_addr
        g0[2] = (uint32_t)ga;                              // global_addr[31:0]
        g0[3] = (uint32_t)((ga >> 32) & 0x1FFFFFFu) | (2u << 30);  // addr[56:32], type=2
        g1[0] = (int)(2u << 16);          // wg_mask=0, data_size=2 (4B)
        g1[1] = (int)(0x800u << 16);      // tensor_dim0 = 2048 (low16 @ [31:16])
        g1[2] = (int)(0x400u << 16);      // td0 hi16=0 | tensor_dim1=1024 low16
        g1[3] = (int)(3u << 16);          // td1 hi16=0 | tile_dim0 = 3 (q)
        g1[4] = (int)(1024u | (3u << 16));// tile_dim1 = 1024 (r) | tile_dim2 = 3 (c)
        g1[5] = (int)2048;                // tensor_dim0_stride = W (low32)
        g1[6] = 0;                        // td0_stride hi16 | td1_stride low16 (0x200000 -> 0)
        g1[7] = 32;                       // td1_stride[47:16] = 0x20 (H*W = 0x200000)
        g2[0] = 3; g2[1] = 0; g2[2] = 0; g2[3] = 0;   // tensor_dim2 = 3 (c)
        g3[0] = 0; g3[1] = 0; g3[2] = 0; g3[3] = 0;
        g4[0] = 0; g4[1] = 0; g4[2] = 0; g4[3] = 0;   // extra group (6-arg form)
        g4[4] = 0; g4[5] = 0; g4[6] = 0; g4[7] = 0;
        __builtin_amdgcn_tensor_load_to_lds(g0, g1, g2, g3, g4, 0);
    }

    // ---- stage bn rows i..i+2, cols 0..257 (overlaps with TDM)
    for (int e = tid; e < 3 * 258; e += 256) {
        int pp = e / 258, col = e - pp * 258;
        bnl[pp * 260 + col] = bn[(i + pp) * WW + col];
    }
    if (__builtin_amdgcn_readfirstlane(wv) == 0)
        __builtin_amdgcn_s_wait_tensorcnt(0);
    __syncthreads();

    // ---- per-lane constant B-matrix (weight^T) registers: K group per A layout
    const int n  = lane & 15;   // WMMA N column / pixel-lane index
    const int hi = lane >> 4;   // 0: K0,K1  1: K2,K3  (16x4 f32 A/B layout)
    float bw0[7], bw1[7];
    #pragma unroll
    for (int s = 0; s < 7; ++s) {
        int k0 = 4 * s + 2 * hi;
        bw0[s] = wB(weight, n, k0);
        bw1[s] = wB(weight, n, k0 + 1);
    }

    float* tr = trans[wv];

    // ---- each wave: 4 tiles of 32 pixels -> columns [wv*128, wv*128+128)
    for (int t4 = 0; t4 < 4; ++t4) {
        const int jbase = wv * 128 + t4 * 32;
        const int jA = jbase + n;        // chain 0 pixel (M = n)
        const int jB = jbase + 16 + n;   // chain 1 pixel
        v8f acc0 = {0.f, 0.f, 0.f, 0.f, 0.f, 0.f, 0.f, 0.f};
        v8f acc1 = {0.f, 0.f, 0.f, 0.f, 0.f, 0.f, 0.f, 0.f};
        #pragma unroll
        for (int s = 0; s < 7; ++s) {
            int k0 = 4 * s + 2 * hi;
            v2f bm; bm[0] = bw0[s]; bm[1] = bw1[s];
            v2f a0; a0[0] = gatherA(bnl, xslab, i, jA, k0);
                    a0[1] = gatherA(bnl, xslab, i, jA, k0 + 1);
            acc0 = __builtin_amdgcn_wmma_f32_16x16x4_f32(
                       false, a0, false, bm, (short)0, acc0, false, false);
            v2f a1; a1[0] = gatherA(bnl, xslab, i, jB, k0);
                    a1[1] = gatherA(bnl, xslab, i, jB, k0 + 1);
            acc1 = __builtin_amdgcn_wmma_f32_16x16x4_f32(
                       false, a1, false, bm, (short)0, acc1, false, false);
        }
        // D layout: vgpr v, lane n+16*(m/8) holds D[m][n], m = hi*8+v
        if (n < 3) {
            #pragma unroll
            for (int v = 0; v < 8; ++v) {
                tr[n * 32 +      hi * 8 + v] = acc0[v];
                tr[n * 32 + 16 + hi * 8 + v] = acc1[v];
            }
        }
        asm volatile("s_wait_dscnt 0x0" ::: "memory");
        // mask + coalesced store (32 consecutive columns per o)
        int jcol = jbase + lane;
        int bnm  = bn[(i + 2) * WW + (jcol + 2)];
        bool ok  = (jcol < WJ) && ((i + 2 + bnm) < HH);
        #pragma unroll
        for (int o = 0; o < 3; ++o) {
            float val = ok ? tr[o * 32 + lane] : 0.0f;
            outB[((size_t)o * HH + i) * WW + jcol] = val;
        }
    }

    // ---- right half j in [1024, 2048): zeros (float4, fully coalesced)
    {
        int col = 1024 + tid * 4;
        #pragma unroll
        for (int o = 0; o < 3; ++o)
            *(float4*)(outB + ((size_t)o * HH + i) * WW + col) =
                make_float4(0.f, 0.f, 0.f, 0.f);
    }
}

extern "C" void kernel_launch(void* const* d_in, const int* in_sizes, int n_in,
                              void* d_out, int out_size, void* d_ws, size_t ws_size,
                              hipStream_t stream) {
    const float* x      = (const float*)d_in[0];
    const float* weight = (const float*)d_in[1];
    const int*   bnm    = (const int*)d_in[2];
    float*       out    = (float*)d_out;
    (void)in_sizes; (void)n_in; (void)out_size; (void)d_ws; (void)ws_size;
    dim3 grid(HH, 4);      // (row i, batch b)
    dim3 block(256);       // 8 wave32
    dis_conv_52243982189252_kernel<<<grid, block, 0, stream>>>(x, weight, bnm, out);
}